// WeightedLoss_61607010893834
// MI455X (gfx1250) — compile-verified
//
#include <hip/hip_runtime.h>
#include <math.h>

// Problem constants (match reference)
#define B_N   8192
#define D_DIM 256     // outputs feature dim
#define C_DIM 128     // labels feature dim
#define TAU   0.5f
#define MAG   1.0f

typedef __attribute__((ext_vector_type(2))) float v2f;
typedef __attribute__((ext_vector_type(4))) float v4f;
typedef __attribute__((ext_vector_type(8))) float v8f;

// LDS row strides, padded by 4 floats so the 16 lanes of a fragment load hit
// 16 distinct banks (stride%64 == 4).
#define OUTP (D_DIM + 4)
#define LABP (C_DIM + 4)

// Monotone float<->uint encoding so we can use integer atomicMin/Max for
// global float min/max reductions.
__device__ __forceinline__ unsigned fenc(float f) {
    unsigned u = __float_as_uint(f);
    return (u & 0x80000000u) ? ~u : (u | 0x80000000u);
}
__device__ __forceinline__ float fdec(unsigned u) {
    u = (u & 0x80000000u) ? (u & 0x7fffffffu) : ~u;
    return __uint_as_float(u);
}

// ---------------------------------------------------------------------------
// Kernel 1: per-row prep. One block (256 threads) per row:
//   sq[row]    = ||outputs[row]||^2
//   labn[row]  = labels[row] / (||labels[row]|| + 1e-12)
//   init stats cells + row-sum buffers
// ---------------------------------------------------------------------------
__global__ __launch_bounds__(256) void k_prep(
    const float* __restrict__ outputs, const float* __restrict__ labels,
    float* __restrict__ labn, float* __restrict__ sq,
    float* __restrict__ posSum, float* __restrict__ negSum,
    unsigned* __restrict__ stats)
{
    const int row = blockIdx.x;
    const int tid = threadIdx.x;

    float o  = outputs[(size_t)row * D_DIM + tid];
    float so = o * o;
    float l  = (tid < C_DIM) ? labels[(size_t)row * C_DIM + tid] : 0.0f;
    float sl = l * l;

    for (int m = 16; m >= 1; m >>= 1) {
        so += __shfl_xor(so, m);
        sl += __shfl_xor(sl, m);
    }
    __shared__ float redO[8], redL[8];
    if ((tid & 31) == 0) { redO[tid >> 5] = so; redL[tid >> 5] = sl; }
    __syncthreads();

    if (tid == 0) {
        float s = 0.f;
        #pragma unroll
        for (int w = 0; w < 8; ++w) s += redO[w];
        sq[row] = s;
        posSum[row] = 0.f;
        negSum[row] = 0.f;
        if (row == 0) {           // encoded-uint atomics init
            stats[0] = 0xFFFFFFFFu;  // sim min  (atomicMin)
            stats[1] = 0u;           // sim max  (atomicMax)
            stats[2] = 0u;           // d2  max  (atomicMax)
        }
    }
    if (tid < C_DIM) {
        float ls = redL[0] + redL[1] + redL[2] + redL[3];
        labn[(size_t)row * C_DIM + tid] = l * (1.0f / (sqrtf(ls) + 1e-12f));
    }
}

// ---------------------------------------------------------------------------
// Kernel 2 (templated): tiled dual-GEMM over the virtual [B,B] matrices.
// Block = 128 threads = 4 waves; each wave owns a 16-row strip and walks all
// 512 column tiles. Per tile: sim via 32 chained V_WMMA_F32_16X16X4_F32
// (K=128) and gram via 64 chained WMMAs (K=256); epilogue fused in registers.
//   MODE 0: reduce min/max(sim), max(d2)  -> stats via atomics
//   MODE 1: normalize with stats, accumulate per-row sum(exp) pos/neg
// ---------------------------------------------------------------------------
template <int MODE>
__global__ __launch_bounds__(128) void k_tiles(
    const float* __restrict__ outputs, const float* __restrict__ labn,
    const float* __restrict__ sq,
    float* __restrict__ posSum, float* __restrict__ negSum,
    unsigned* __restrict__ stats)
{
    __shared__ float sOut[4][16 * OUTP];
    __shared__ float sLab[4][16 * LABP];

    const int tid   = threadIdx.x;
    const int wave  = tid >> 5;
    const int lane  = tid & 31;
    const int lrow  = lane & 15;        // M for A-frag / N for B-frag
    const int khalf = (lane >> 4) * 2;  // per-lane K offset inside a K=4 step

    const int i0 = (blockIdx.x * 4 + wave) * 16;  // this wave's 16 rows

    // Stage this wave's 16 A-rows into padded LDS (float4 copies, aligned).
    float* aOut = sOut[wave];
    float* aLab = sLab[wave];
    for (int idx = lane; idx < 16 * (D_DIM / 4); idx += 32) {
        int r = idx >> 6, c4 = idx & 63;
        v4f v = *(const v4f*)&outputs[(size_t)(i0 + r) * D_DIM + c4 * 4];
        *(v4f*)&aOut[r * OUTP + c4 * 4] = v;
    }
    for (int idx = lane; idx < 16 * (C_DIM / 4); idx += 32) {
        int r = idx >> 5, c4 = idx & 31;
        v4f v = *(const v4f*)&labn[(size_t)(i0 + r) * C_DIM + c4 * 4];
        *(v4f*)&aLab[r * LABP + c4 * 4] = v;
    }
    __syncthreads();

    float smin = 0.f, invSR = 1.f, invEmax = 1.f;
    if (MODE == 1) {
        smin        = fdec(stats[0]);
        float smax  = fdec(stats[1]);
        float d2max = fdec(stats[2]);
        invSR   = 1.0f / (smax - smin);
        invEmax = (d2max > 0.f) ? (1.0f / sqrtf(d2max)) : 1.0f;
    }

    // ||x_i||^2 for this lane's 8 output rows (C-fragment row mapping).
    float sqi[8];
    #pragma unroll
    for (int r = 0; r < 8; ++r) sqi[r] = sq[i0 + r + 8 * (lane >> 4)];

    float posAcc[8], negAcc[8];
    #pragma unroll
    for (int r = 0; r < 8; ++r) { posAcc[r] = 0.f; negAcc[r] = 0.f; }
    float lmin = 3.4e38f, lmax = -3.4e38f, ld2max = 0.f;

    for (int j0 = 0; j0 < B_N; j0 += 16) {
        // --- cosine-sim tile: labn_i . labn_j, K = 128 ---
        v8f cs = {};
        {
            const float* bl = &labn[(size_t)(j0 + lrow) * C_DIM + khalf];
            const float* al = &aLab[lrow * LABP + khalf];
            #pragma unroll
            for (int kb = 0; kb < C_DIM; kb += 4) {
                v2f a = *(const v2f*)&al[kb];
                v2f b = *(const v2f*)&bl[kb];
                cs = __builtin_amdgcn_wmma_f32_16x16x4_f32(
                        false, a, false, b, (short)0, cs, false, false);
            }
        }
        // --- gram tile: out_i . out_j, K = 256 ---
        v8f cg = {};
        {
            const float* bo = &outputs[(size_t)(j0 + lrow) * D_DIM + khalf];
            const float* ao = &aOut[lrow * OUTP + khalf];
            #pragma unroll
            for (int kb = 0; kb < D_DIM; kb += 4) {
                v2f a = *(const v2f*)&ao[kb];
                v2f b = *(const v2f*)&bo[kb];
                cg = __builtin_amdgcn_wmma_f32_16x16x4_f32(
                        false, a, false, b, (short)0, cg, false, false);
            }
        }

        float sqj = sq[j0 + lrow];  // ||x_j||^2 for this lane's column
        #pragma unroll
        for (int r = 0; r < 8; ++r) {
            float s_raw = cs[r];
            float d2    = sqi[r] + sqj - 2.0f * cg[r];
            if (MODE == 0) {
                lmin   = fminf(lmin, s_raw);
                lmax   = fmaxf(lmax, s_raw);
                ld2max = fmaxf(ld2max, d2);
            } else {
                float s    = (s_raw - smin) * invSR;
                float eud  = (d2 > 0.f) ? sqrtf(d2) * invEmax : 0.f;
                float dist = eud + s;
                if (s > TAU) posAcc[r] += __expf(dist);
                else         negAcc[r] += __expf(MAG - dist);
            }
        }
    }

    if (MODE == 0) {
        for (int m = 16; m >= 1; m >>= 1) {
            lmin   = fminf(lmin,   __shfl_xor(lmin, m));
            lmax   = fmaxf(lmax,   __shfl_xor(lmax, m));
            ld2max = fmaxf(ld2max, __shfl_xor(ld2max, m));
        }
        if (lane == 0) {
            atomicMin(&stats[0], fenc(lmin));
            atomicMax(&stats[1], fenc(lmax));
            atomicMax(&stats[2], fenc(fmaxf(ld2max, 0.f)));
        }
    } else {
        // Row sums: each row lives across one 16-lane half; butterfly-reduce.
        #pragma unroll
        for (int r = 0; r < 8; ++r) {
            for (int m = 1; m < 16; m <<= 1) {
                posAcc[r] += __shfl_xor(posAcc[r], m);
                negAcc[r] += __shfl_xor(negAcc[r], m);
            }
        }
        if (lrow == 0) {  // lanes 0 and 16 each own 8 rows; exclusive writes
            int rbase = i0 + 8 * (lane >> 4);
            #pragma unroll
            for (int r = 0; r < 8; ++r) {
                posSum[rbase + r] = posAcc[r];
                negSum[rbase + r] = negAcc[r];
            }
        }
    }
}

// ---------------------------------------------------------------------------
// Kernel 3: final scalar.  loss = mean(max(log posSum,0) + max(log negSum,0))
// (negSum may be 0 -> log = -inf -> clamped to 0, matching the reference).
// ---------------------------------------------------------------------------
__global__ __launch_bounds__(256) void k_final(
    const float* __restrict__ posSum, const float* __restrict__ negSum,
    float* __restrict__ out)
{
    __shared__ float red[8];
    const int tid = threadIdx.x;
    float acc = 0.f;
    for (int i = tid; i < B_N; i += 256) {
        acc += fmaxf(logf(posSum[i]), 0.f) + fmaxf(logf(negSum[i]), 0.f);
    }
    for (int m = 16; m >= 1; m >>= 1) acc += __shfl_xor(acc, m);
    if ((tid & 31) == 0) red[tid >> 5] = acc;
    __syncthreads();
    if (tid == 0) {
        float t = 0.f;
        #pragma unroll
        for (int w = 0; w < 8; ++w) t += red[w];
        out[0] = t / (float)B_N;
    }
}

// ---------------------------------------------------------------------------
extern "C" void kernel_launch(void* const* d_in, const int* in_sizes, int n_in,
                              void* d_out, int out_size, void* d_ws, size_t ws_size,
                              hipStream_t stream) {
    const float* outputs = (const float*)d_in[0];  // [8192, 256] f32
    const float* labels  = (const float*)d_in[1];  // [8192, 128] f32

    // Workspace layout (~4.3 MB):
    char*  ws     = (char*)d_ws;
    float* labn   = (float*)ws;                                 // 8192*128
    float* sq     = (float*)(ws + (size_t)B_N * C_DIM * sizeof(float));
    float* posSum = sq + B_N;
    float* negSum = posSum + B_N;
    unsigned* stats = (unsigned*)(negSum + B_N);

    k_prep<<<B_N, 256, 0, stream>>>(outputs, labels, labn, sq, posSum, negSum, stats);
    k_tiles<0><<<B_N / 64, 128, 0, stream>>>(outputs, labn, sq, posSum, negSum, stats);
    k_tiles<1><<<B_N / 64, 128, 0, stream>>>(outputs, labn, sq, posSum, negSum, stats);
    k_final<<<1, 256, 0, stream>>>(posSum, negSum, (float*)d_out);
}